// Dimension_34187939676165
// MI455X (gfx1250) — compile-verified
//
#include <hip/hip_runtime.h>
#include <math.h>

typedef __attribute__((ext_vector_type(2))) float v2f;
typedef __attribute__((ext_vector_type(8))) float v8f;

#define INFF __builtin_inff()

__device__ __forceinline__ void merge2(float& a1, float& a2, float b1, float b2) {
    float lo = fminf(a1, b1);
    float hi = fmaxf(a1, b1);
    a2 = fminf(fminf(a2, b2), hi);
    a1 = lo;
}

// Fused epilogue: consume one 16x16 Gram tile, update per-lane two-smallest
// of (sq_col - 2*G) for the 8 C-rows this lane holds.
__device__ __forceinline__ void tile_top2(const v8f& c, int jt, int tile,
                                          int l16, int half, int n,
                                          const float* __restrict__ sq,
                                          float* __restrict__ m1,
                                          float* __restrict__ m2) {
    const int col = jt * 16 + l16;
    const float sc = sq[col];  // sq padded to ntiles*16; pad cols masked below
#pragma unroll
    for (int v = 0; v < 8; ++v) {
        const int row = tile * 16 + v + 8 * half;
        float t = sc - 2.0f * c[v];
        if (col >= n || col == row) t = INFF;  // mask pad columns + diagonal
        float hi = fmaxf(m1[v], t);
        m1[v] = fminf(m1[v], t);
        m2[v] = fminf(m2[v], hi);
    }
}

__device__ __forceinline__ void finish_rows(float* m1, float* m2, int tile,
                                            int l16, int half, int n, int b,
                                            const float* __restrict__ sq,
                                            float* __restrict__ xlog) {
    // reduce two-smallest across the 16 lanes of each half-wave
#pragma unroll
    for (int v = 0; v < 8; ++v) {
        for (int off = 8; off > 0; off >>= 1) {
            float o1 = __shfl_xor(m1[v], off);
            float o2 = __shfl_xor(m2[v], off);
            merge2(m1[v], m2[v], o1, o2);
        }
    }
    if (l16 == 0) {  // lanes 0 / 16 own rows v / v+8
#pragma unroll
        for (int v = 0; v < 8; ++v) {
            const int row = tile * 16 + v + 8 * half;
            if (row < n) {
                const float sr  = sq[row];
                const float d2a = fmaxf(sr + m1[v], 0.0f);
                const float d2b = fmaxf(sr + m2[v], 0.0f);
                // log(sqrt(d2b)/sqrt(d2a))
                xlog[(size_t)b * n + row] = 0.5f * (logf(d2b) - logf(d2a));
            }
        }
    }
}

// ---------------------------------------------------------------------------
// Phase 1 (fast path): X staged zero-padded in LDS; A fragments hoisted to
// registers; inner loop = ds_load_b64 (B) + v_wmma_f32_16x16x4_f32, branch-free.
// Grid (B, rowTileGroups), 256 threads = 8 wave32s; wave w owns one row tile.
// ---------------------------------------------------------------------------
__global__ __launch_bounds__(256) void twonn_dist_lds(
    const float* __restrict__ X, float* __restrict__ xlog,
    int n, int d, int ntiles, int dpad) {
    extern __shared__ float smem[];
    const int npad = ntiles * 16;
    float* xs = smem;                          // npad * dpad, zero-padded X
    float* sq = smem + (size_t)npad * dpad;    // npad squared norms (pad = 0)

    const int b = blockIdx.x;
    const float* Xb = X + (size_t)b * (size_t)n * (size_t)d;

    for (int i = threadIdx.x; i < npad * dpad; i += blockDim.x) {
        const int row = i / dpad;
        const int k   = i - row * dpad;
        xs[i] = (row < n && k < d) ? Xb[(size_t)row * d + k] : 0.f;
    }
    __syncthreads();
    for (int i = threadIdx.x; i < npad; i += blockDim.x) {
        const float* r = xs + (size_t)i * dpad;
        float s = 0.f;
        for (int k = 0; k < dpad; ++k) s += r[k] * r[k];
        sq[i] = s;
    }
    __syncthreads();

    const int lane   = threadIdx.x & 31;
    const int wave   = threadIdx.x >> 5;
    const int nwaves = blockDim.x >> 5;
    const int l16    = lane & 15;
    const int half   = lane >> 4;

    const int tile = blockIdx.y * nwaves + wave;
    if (tile >= ntiles) return;  // uniform per wave; EXEC stays all-1s

    const int kch = dpad >> 2;   // K-chunks of 4
    const float* arow = xs + (size_t)(tile * 16 + l16) * dpad + 2 * half;

    float m1[8], m2[8];
#pragma unroll
    for (int v = 0; v < 8; ++v) { m1[v] = INFF; m2[v] = INFF; }

    if (kch <= 8) {  // d <= 32: A fragments live in registers
        v2f areg[8];
#pragma unroll
        for (int kc = 0; kc < 8; ++kc)
            if (kc < kch) areg[kc] = *(const v2f*)(arow + 4 * kc);
        for (int jt = 0; jt < ntiles; ++jt) {
            const float* brow = xs + (size_t)(jt * 16 + l16) * dpad + 2 * half;
            v8f c = {};
#pragma unroll
            for (int kc = 0; kc < 8; ++kc) {
                if (kc < kch) {
                    v2f bv = *(const v2f*)(brow + 4 * kc);
                    c = __builtin_amdgcn_wmma_f32_16x16x4_f32(
                            false, areg[kc], false, bv, (short)0, c, false, false);
                }
            }
            tile_top2(c, jt, tile, l16, half, n, sq, m1, m2);
        }
    } else {  // generic large-d path: both fragments streamed from LDS
        for (int jt = 0; jt < ntiles; ++jt) {
            const float* brow = xs + (size_t)(jt * 16 + l16) * dpad + 2 * half;
            v8f c = {};
            for (int kc = 0; kc < kch; ++kc) {
                v2f av = *(const v2f*)(arow + 4 * kc);
                v2f bv = *(const v2f*)(brow + 4 * kc);
                c = __builtin_amdgcn_wmma_f32_16x16x4_f32(
                        false, av, false, bv, (short)0, c, false, false);
            }
            tile_top2(c, jt, tile, l16, half, n, sq, m1, m2);
        }
    }

    finish_rows(m1, m2, tile, l16, half, n, b, sq, xlog);
}

// ---------------------------------------------------------------------------
// Phase 1 fallback for X too large for LDS: guarded global loads (slow path).
// ---------------------------------------------------------------------------
__global__ __launch_bounds__(256) void twonn_dist_global(
    const float* __restrict__ X, float* __restrict__ xlog,
    int n, int d, int ntiles) {
    extern __shared__ float sq[];  // n floats
    const int b = blockIdx.x;
    const float* Xb = X + (size_t)b * (size_t)n * (size_t)d;

    for (int i = threadIdx.x; i < n; i += blockDim.x) {
        const float* row = Xb + (size_t)i * d;
        float s = 0.f;
        for (int k = 0; k < d; ++k) { float v = row[k]; s += v * v; }
        sq[i] = s;
    }
    __syncthreads();

    const int lane   = threadIdx.x & 31;
    const int wave   = threadIdx.x >> 5;
    const int nwaves = blockDim.x >> 5;
    const int l16    = lane & 15;
    const int half   = lane >> 4;

    const int tile = blockIdx.y * nwaves + wave;
    if (tile >= ntiles) return;

    const int rowLoad = tile * 16 + l16;
    float m1[8], m2[8];
#pragma unroll
    for (int v = 0; v < 8; ++v) { m1[v] = INFF; m2[v] = INFF; }

    for (int jt = 0; jt < ntiles; ++jt) {
        const int colLoad = jt * 16 + l16;
        v8f c = {};
        for (int kk = 0; kk < d; kk += 4) {
            const int k0 = kk + 2 * half;
            v2f av, bv;
            av.x = (rowLoad < n && k0     < d) ? Xb[(size_t)rowLoad * d + k0    ] : 0.f;
            av.y = (rowLoad < n && k0 + 1 < d) ? Xb[(size_t)rowLoad * d + k0 + 1] : 0.f;
            bv.x = (colLoad < n && k0     < d) ? Xb[(size_t)colLoad * d + k0    ] : 0.f;
            bv.y = (colLoad < n && k0 + 1 < d) ? Xb[(size_t)colLoad * d + k0 + 1] : 0.f;
            c = __builtin_amdgcn_wmma_f32_16x16x4_f32(
                    false, av, false, bv, (short)0, c, false, false);
        }
        const int col = jt * 16 + l16;
        const float sc = (col < n) ? sq[col] : 0.f;
#pragma unroll
        for (int v = 0; v < 8; ++v) {
            const int row = tile * 16 + v + 8 * half;
            float t = sc - 2.0f * c[v];
            if (col >= n || col == row) t = INFF;
            float hi = fmaxf(m1[v], t);
            m1[v] = fminf(m1[v], t);
            m2[v] = fminf(m2[v], hi);
        }
    }
    finish_rows(m1, m2, tile, l16, half, n, b, sq, xlog);
}

// ---------------------------------------------------------------------------
// Phase 2: result_b = sum(x_sorted*y)/sum(x*x), y_j = -log(1-j/n), via rank
// counting in LDS (no explicit sort). One block per batch.
// ---------------------------------------------------------------------------
__global__ __launch_bounds__(256) void twonn_fit_kernel(
    const float* __restrict__ xlog, float* __restrict__ out, int n) {
    extern __shared__ float xs[];
    __shared__ float redxy[8];
    __shared__ float redxx[8];
    const int b = blockIdx.x;
    const float* xb = xlog + (size_t)b * n;
    for (int i = threadIdx.x; i < n; i += blockDim.x) xs[i] = xb[i];
    __syncthreads();

    float sxy = 0.f, sxx = 0.f;
    const float fn = (float)n;
    for (int i = threadIdx.x; i < n; i += blockDim.x) {
        const float xi = xs[i];
        int r = 0;
        for (int j = 0; j < n; ++j) {
            const float xj = xs[j];
            r += (int)((xj < xi) || (xj == xi && j < i));
        }
        const float y = logf(fn / (float)(n - r));
        sxy += xi * y;
        sxx += xi * xi;
    }
    for (int off = 16; off > 0; off >>= 1) {
        sxy += __shfl_xor(sxy, off);
        sxx += __shfl_xor(sxx, off);
    }
    const int lane = threadIdx.x & 31, wave = threadIdx.x >> 5;
    if (lane == 0) { redxy[wave] = sxy; redxx[wave] = sxx; }
    __syncthreads();
    if (threadIdx.x == 0) {
        float a = 0.f, c = 0.f;
        const int nw = blockDim.x >> 5;
        for (int w = 0; w < nw; ++w) { a += redxy[w]; c += redxx[w]; }
        out[b] = a / c;
    }
}

extern "C" void kernel_launch(void* const* d_in, const int* in_sizes, int n_in,
                              void* d_out, int out_size, void* d_ws, size_t ws_size,
                              hipStream_t stream) {
    const float* X = (const float*)d_in[0];
    float* out = (float*)d_out;

    const long long total = (long long)in_sizes[0];
    const int B = (out_size > 0) ? out_size : 1;   // output is [B]
    const long long nd = total / (long long)B;

    // Split n*d: prefer modest ambient dimension (typical TwoNN inputs).
    static const int dcand[] = {8, 16, 4, 32, 2, 64, 3, 128, 5, 6, 12, 24,
                                10, 20, 48, 96, 256, 7, 9, 512};
    int d = 1;
    for (int i = 0; i < (int)(sizeof(dcand) / sizeof(int)); ++i) {
        if (nd % dcand[i] == 0 && (nd / dcand[i]) >= 16) { d = dcand[i]; break; }
    }
    const int n = (int)(nd / d);

    float* xlog = (float*)d_ws;            // B*n floats
    const int ntiles = (n + 15) / 16;
    const int npad   = ntiles * 16;
    const int dpad   = (d + 3) & ~3;
    const int nwaves = 8;                  // 256 threads
    dim3 grid1((unsigned)B, (unsigned)((ntiles + nwaves - 1) / nwaves));

    const size_t smem_lds = ((size_t)npad * dpad + npad) * sizeof(float);
    if (smem_lds <= (size_t)(160 * 1024)) {
        twonn_dist_lds<<<grid1, 256, smem_lds, stream>>>(X, xlog, n, d, ntiles, dpad);
    } else {
        twonn_dist_global<<<grid1, 256, (size_t)n * sizeof(float), stream>>>(
            X, xlog, n, d, ntiles);
    }
    twonn_fit_kernel<<<B, 256, (size_t)n * sizeof(float), stream>>>(xlog, out, n);
}